// RegimeRulesNet_87531433492893
// MI455X (gfx1250) — compile-verified
//
#include <hip/hip_runtime.h>
#include <math.h>

typedef __attribute__((ext_vector_type(2))) float v2f;
typedef __attribute__((ext_vector_type(8))) float v8f;

#define LOG2E_F 1.44269504088896340736f
#define LN2_F   0.69314718055994530942f

constexpr int B_ = 2048, D_ = 256, K_ = 6, R_ = 256, G_ = 16, S_ = 16;
constexpr int TB = 16;           // batch tile (== WMMA M)
constexpr int RH = 128;          // rules per block (R split across gridDim.y = 2)
constexpr float BETA_GROUP = 6.f, BETA_K = 8.f, BETA_RULES = 6.f;

// workspace layout (float offsets)
constexpr size_t OFF_A2    = 0;
constexpr size_t OFF_B2    = OFF_A2 + (size_t)R_ * D_;
constexpr size_t OFF_W2    = OFF_B2 + (size_t)R_ * D_;
constexpr size_t OFF_W2SUM = OFF_W2 + (size_t)R_ * D_;
constexpr size_t OFF_A1    = OFF_W2SUM + R_;
constexpr size_t OFF_B1    = OFF_A1 + (size_t)K_ * D_;
constexpr size_t OFF_W1    = OFF_B1 + (size_t)K_ * D_;
constexpr size_t OFF_PART  = OFF_W1 + (size_t)K_ * D_;
// total = 3*65536 + 256 + 3*1536 + 2*2048 floats ~= 803 KB

__device__ __forceinline__ float fexp2(float x) { return __builtin_amdgcn_exp2f(x); }
__device__ __forceinline__ float frcp(float x)  { return __builtin_amdgcn_rcpf(x); }
// sigmoid(v) = 1/(1+exp(-v)) = rcp(1 + exp2(-log2e*v))   -> v_exp_f32 + v_rcp_f32
__device__ __forceinline__ float sigm(float v)  { return frcp(1.f + fexp2(-LOG2E_F * v)); }

// CDNA5 async memory->LDS copy (ASYNCcnt path, no VGPR data round-trip).
// vdst = LDS byte address (low 32 bits of generic shared pointer, per FLAT
// aperture truncation rule), vaddr = 64-bit global address. 16B aligned.
__device__ __forceinline__ void async_ld_b128(uint32_t lds_addr, const void* gaddr) {
  asm volatile("global_load_async_to_lds_b128 %0, %1, off"
               :: "v"(lds_addr), "v"((uint64_t)(uintptr_t)gaddr)
               : "memory");
}
__device__ __forceinline__ void wait_async0() {
  asm volatile("s_wait_asynccnt 0" ::: "memory");
}

// ---------------- prep: fold tanh/sigmoid/kappa/log2e into per-(r,d) A,B,w ----------------
__global__ void prep_params(const float* __restrict__ th_r, const float* __restrict__ ineq_r,
                            const float* __restrict__ esign_r, const float* __restrict__ mask_r,
                            const float* __restrict__ lk_r,
                            const float* __restrict__ th,  const float* __restrict__ ineq,
                            const float* __restrict__ es,  const float* __restrict__ ml,
                            const float* __restrict__ lk2, float* __restrict__ ws) {
  int i = blockIdx.x * blockDim.x + threadIdx.x;
  if (i < R_ * D_) {
    float kap2 = fminf(fmaxf(expf(lk2[0]), 0.5f), 50.f);
    // c = sigmoid(kap*tanh(ineq)*(x-th)) = rcp(1 + exp2(A*x + B)),  A = -log2e*kap*tanh(ineq)
    float A = -LOG2E_F * kap2 * tanhf(ineq[i]);
    ws[OFF_A2 + i] = A;
    ws[OFF_B2 + i] = -A * th[i];
    ws[OFF_W2 + i] = (1.f / (1.f + expf(-ml[i]))) * tanhf(es[i]);
  }
  if (i < K_ * D_) {
    float kap1 = fminf(fmaxf(expf(lk_r[0]), 0.5f), 50.f);
    float A = -LOG2E_F * kap1 * tanhf(ineq_r[i]);
    ws[OFF_A1 + i] = A;
    ws[OFF_B1 + i] = -A * th_r[i];
    ws[OFF_W1 + i] = (1.f / (1.f + expf(-mask_r[i]))) * tanhf(esign_r[i]);
  }
}

// w2sum[r] = sum_d w2[r,d]   (lets hot loop use  ev = 2*sum(w*c) - w2sum)
__global__ void prep_w2sum(float* __restrict__ ws) {
  int r = threadIdx.x;
  float s = 0.f;
  for (int d = 0; d < D_; ++d) s += ws[OFF_W2 + (size_t)r * D_ + d];
  ws[OFF_W2SUM + r] = s;
}

// ---------------- main: one block = 16 batches x 128 rules (gridDim.y splits R) ----------------
__global__ __launch_bounds__(256)
void regime_rules_main(const float* __restrict__ x,
                       const float* __restrict__ gate_tau, const float* __restrict__ gate_k,
                       const float* __restrict__ prior,    const float* __restrict__ tvec,
                       const float* __restrict__ W,        const float* __restrict__ bco,
                       const int*   __restrict__ gidx,
                       const float* __restrict__ ws,       float* __restrict__ partial) {
  __shared__ __align__(16) float xs[TB * D_];   // 16 KB  x tile
  __shared__ float wt[RH * 16];     //  8 KB  W^T chunk, padded N=16
  __shared__ float part[RH * TB];   //  8 KB  d-half partial accumulators
  __shared__ float zb[RH * TB];     //  8 KB  z_rules[r_local][b]
  __shared__ float lg[TB * 8];      //        regime logits
  __shared__ float pis[TB * 16];    //        softmax pi, padded to 16

  const int tid = threadIdx.x;
  const int bx  = blockIdx.x;       // batch tile
  const int ry  = blockIdx.y;       // which half of R
  const int b0  = bx * TB;

  { // stage x tile via CDNA5 async memory->LDS (ASYNCcnt), 16B chunks
    const char* xg = (const char*)(x + (size_t)b0 * D_);
    const uint32_t xl = (uint32_t)(uintptr_t)(void*)xs;
    for (int i = tid; i < TB * D_ / 4; i += 256)
      async_ld_b128(xl + (uint32_t)i * 16u, xg + (size_t)i * 16);
    wait_async0();
  }
  // stage W^T chunk: wt[rl][n] = W[n, ry*RH + rl] (0-padded for n >= 6)
  for (int i = tid; i < RH * 16; i += 256) {
    int rl = i >> 4, n = i & 15;
    wt[i] = (n < K_) ? W[(size_t)n * R_ + ry * RH + rl] : 0.f;
  }
  __syncthreads();

  // ---- rules accumulation: thread = (rule r, d-half), 16 batch accumulators ----
  const int rl = tid & (RH - 1);
  const int dh = tid >> 7;                 // wave-uniform (waves 0-3: 0, waves 4-7: 1)
  const int r  = ry * RH + rl;
  const int d0 = dh * 128;

  float acc[TB];
  #pragma unroll
  for (int b = 0; b < TB; ++b) acc[b] = 0.f;

  const float4* pa = (const float4*)(ws + OFF_A2 + (size_t)r * D_ + d0);
  const float4* pb = (const float4*)(ws + OFF_B2 + (size_t)r * D_ + d0);
  const float4* pw = (const float4*)(ws + OFF_W2 + (size_t)r * D_ + d0);

  for (int q = 0; q < 32; ++q) {           // 128 d's in float4 steps
    float4 a4 = pa[q], b4 = pb[q], w4 = pw[q];
    #pragma unroll
    for (int b = 0; b < TB; ++b) {
      float4 xv = *(const float4*)&xs[b * D_ + d0 + q * 4];   // wave-uniform LDS broadcast
      float c0 = frcp(1.f + fexp2(fmaf(a4.x, xv.x, b4.x)));
      float c1 = frcp(1.f + fexp2(fmaf(a4.y, xv.y, b4.y)));
      float c2 = frcp(1.f + fexp2(fmaf(a4.z, xv.z, b4.z)));
      float c3 = frcp(1.f + fexp2(fmaf(a4.w, xv.w, b4.w)));
      float s01 = fmaf(w4.x, c0, w4.y * c1);
      float s23 = fmaf(w4.z, c2, w4.w * c3);
      acc[b] += s01 + s23;
    }
  }
  if (dh == 1) {
    #pragma unroll
    for (int b = 0; b < TB; ++b) part[rl * TB + b] = acc[b];
  }
  __syncthreads();

  if (dh == 0) {
    // combine halves -> z_rules
    float w2s = ws[OFF_W2SUM + r];
    float tr  = tvec[r];
    #pragma unroll
    for (int b = 0; b < TB; ++b) {
      float ev = 2.f * (acc[b] + part[rl * TB + b]) - w2s;   // = sum_d w*(2c-1)
      zb[rl * TB + b] = sigm(BETA_RULES * (ev - tr));
    }
  } else {
    // regime path (cheap, duplicated in both ry blocks): thread = (batch, k)
    int idx = tid - 128;
    if (idx < TB * K_) {
      int b = idx / K_, k = idx % K_;
      const float* A1 = ws + OFF_A1 + (size_t)k * D_;
      const float* B1 = ws + OFF_B1 + (size_t)k * D_;
      const float* w1 = ws + OFF_W1 + (size_t)k * D_;
      float ssum = 0.f;
      for (int g = 0; g < G_; ++g) {
        float eg = 0.f;
        for (int s2 = 0; s2 < S_; ++s2) {
          int d = gidx[g * S_ + s2];
          float c = frcp(1.f + fexp2(fmaf(A1[d], xs[b * D_ + d], B1[d])));
          eg = fmaf(w1[d], fmaf(2.f, c, -1.f), eg);
        }
        ssum += sigm(BETA_GROUP * (eg - gate_tau[k * G_ + g]));
      }
      float keff = (float)G_ * sigm(gate_k[k]);
      float zr = fmaxf(sigm(BETA_K * (ssum - keff)), 1e-6f);
      lg[b * 8 + k] = prior[k] + __builtin_amdgcn_logf(zr) * LN2_F;  // v_log_f32 is log2
    }
  }
  __syncthreads();

  if (tid < TB) { // softmax over K=6 per batch
    int b = tid;
    float m = -1e30f;
    for (int k = 0; k < K_; ++k) m = fmaxf(m, lg[b * 8 + k]);
    float e[K_], sum = 0.f;
    for (int k = 0; k < K_; ++k) { e[k] = fexp2((lg[b * 8 + k] - m) * LOG2E_F); sum += e[k]; }
    float inv = frcp(sum);
    for (int k = 0; k < 16; ++k) pis[b * 16 + k] = (k < K_) ? e[k] * inv : 0.f;
  }
  __syncthreads();

  // ---- combine: yk = z @ W^T via V_WMMA_F32_16X16X4_F32, wave 0 only (EXEC all ones) ----
  if (tid < 32) {
    const int lane = tid;
    const int half = lane >> 4;   // 0: M rows 0-7 / K 0-1 ; 1: M rows 8-15 / K 2-3
    const int ln   = lane & 15;
    v8f c;
    float cinit = (ry == 0 && ln < K_) ? bco[ln] : 0.f;   // +b added once (ry==0)
    #pragma unroll
    for (int j = 0; j < 8; ++j) c[j] = cinit;
    for (int s = 0; s < RH / 4; ++s) {
      int r0 = s * 4 + 2 * half;
      v2f a, bf;
      a.x  = zb[(r0    ) * TB + ln];  a.y  = zb[(r0 + 1) * TB + ln];   // A[M=ln, K]
      bf.x = wt[(r0    ) * 16 + ln];  bf.y = wt[(r0 + 1) * 16 + ln];   // B[K, N=ln]
      c = __builtin_amdgcn_wmma_f32_16x16x4_f32(false, a, false, bf, (short)0, c,
                                                false, false);
    }
    // out_partial[b] = sum_n pi[b,n] * yk[b,n] : lane-n weighted, reduce over 16 lanes
    #pragma unroll
    for (int j = 0; j < 8; ++j) {
      int m = j + 8 * half;                         // batch row in tile
      float val = c[j] * pis[m * 16 + ln];
      val += __shfl_xor(val, 8, 32);
      val += __shfl_xor(val, 4, 32);
      val += __shfl_xor(val, 2, 32);
      val += __shfl_xor(val, 1, 32);
      if (ln == 0) partial[(size_t)ry * B_ + b0 + m] = val;
    }
  }
}

__global__ void finalize_out(const float* __restrict__ partial,
                             const float* __restrict__ bias, float* __restrict__ out) {
  int i = blockIdx.x * blockDim.x + threadIdx.x;
  if (i < B_) out[i] = partial[i] + partial[B_ + i] + bias[0];
}

extern "C" void kernel_launch(void* const* d_in, const int* in_sizes, int n_in,
                              void* d_out, int out_size, void* d_ws, size_t ws_size,
                              hipStream_t stream) {
  const float* x        = (const float*)d_in[0];
  const float* th_r     = (const float*)d_in[1];
  const float* ineq_r   = (const float*)d_in[2];
  const float* esign_r  = (const float*)d_in[3];
  const float* mask_r   = (const float*)d_in[4];
  const float* lk_r     = (const float*)d_in[5];
  const float* gate_tau = (const float*)d_in[6];
  const float* gate_k   = (const float*)d_in[7];
  const float* prior    = (const float*)d_in[8];
  const float* th       = (const float*)d_in[9];
  const float* ineq     = (const float*)d_in[10];
  const float* e_sign   = (const float*)d_in[11];
  const float* ml       = (const float*)d_in[12];
  const float* lk2      = (const float*)d_in[13];
  const float* tvec     = (const float*)d_in[14];
  const float* W        = (const float*)d_in[15];
  const float* bco      = (const float*)d_in[16];
  const float* bias     = (const float*)d_in[17];
  const int*   gidx     = (const int*)d_in[18];
  float* wsf = (float*)d_ws;
  float* out = (float*)d_out;

  prep_params<<<(R_ * D_ + 255) / 256, 256, 0, stream>>>(th_r, ineq_r, esign_r, mask_r, lk_r,
                                                         th, ineq, e_sign, ml, lk2, wsf);
  prep_w2sum<<<1, 256, 0, stream>>>(wsf);
  regime_rules_main<<<dim3(B_ / TB, 2), 256, 0, stream>>>(x, gate_tau, gate_k, prior, tvec,
                                                          W, bco, gidx, wsf, wsf + OFF_PART);
  finalize_out<<<B_ / 256, 256, 0, stream>>>(wsf + OFF_PART, bias, out);
}